// TripletLossWithHardMining_10264971838200
// MI455X (gfx1250) — compile-verified
//
#include <hip/hip_runtime.h>
#include <math.h>

typedef __attribute__((ext_vector_type(2))) float v2f;
typedef __attribute__((ext_vector_type(8))) float v8f;

#define BDIM 4096
#define DDIM 1024
#define EPSV 1e-6f

// ---------------------------------------------------------------------------
// Phase 1: per-row statistics + init of hard-mining accumulators.
// One wave (32 lanes) per row, 8 rows per 256-thread block.
// ws layout (floats):
//   [0*B) rowTermA   = ||a||^2 + 2*eps*sum(a) + D*eps^2
//   [1*B) colTermN   = ||n||^2 - 2*eps*sum(n)
//   [2*B) colTermP   = ||p||^2 - 2*eps*sum(p)
//   [3*B) d_ap       = ||a - p + eps||
//   [4*B) d_an_row   = ||a - n + eps||
//   [5*B) hardN (uint bits, init 0.0f,  atomicMax)
//   [6*B) hardP (uint bits, init +inf,  atomicMin)
// ---------------------------------------------------------------------------
__global__ __launch_bounds__(256) void rowstats_kernel(
    const float* __restrict__ A, const float* __restrict__ P,
    const float* __restrict__ N, float* __restrict__ ws) {
  int wv = threadIdx.x >> 5;
  int lane = threadIdx.x & 31;
  int row = blockIdx.x * 8 + wv;
  const float4* a4 = (const float4*)(A + (size_t)row * DDIM);
  const float4* p4 = (const float4*)(P + (size_t)row * DDIM);
  const float4* n4 = (const float4*)(N + (size_t)row * DDIM);

  float sa = 0.f, na2 = 0.f, sp = 0.f, np2 = 0.f, sn = 0.f, nn2 = 0.f;
  float dap2 = 0.f, dan2 = 0.f;
#pragma unroll
  for (int t = 0; t < 8; ++t) {
    int idx = lane + 32 * t;  // 256 float4 per row
    float4 av = a4[idx];
    float4 pv = p4[idx];
    float4 nv = n4[idx];
    const float* ap = (const float*)&av;
    const float* pp = (const float*)&pv;
    const float* np = (const float*)&nv;
#pragma unroll
    for (int c = 0; c < 4; ++c) {
      float a = ap[c], p = pp[c], n = np[c];
      sa += a;  na2 += a * a;
      sp += p;  np2 += p * p;
      sn += n;  nn2 += n * n;
      float dp = a - p + EPSV;  dap2 += dp * dp;
      float dn = a - n + EPSV;  dan2 += dn * dn;
    }
  }
#pragma unroll
  for (int m = 16; m >= 1; m >>= 1) {
    sa   += __shfl_xor(sa, m, 32);
    na2  += __shfl_xor(na2, m, 32);
    sp   += __shfl_xor(sp, m, 32);
    np2  += __shfl_xor(np2, m, 32);
    sn   += __shfl_xor(sn, m, 32);
    nn2  += __shfl_xor(nn2, m, 32);
    dap2 += __shfl_xor(dap2, m, 32);
    dan2 += __shfl_xor(dan2, m, 32);
  }
  if (lane == 0) {
    ws[0 * BDIM + row] = na2 + 2.0f * EPSV * sa + (float)DDIM * EPSV * EPSV;
    ws[1 * BDIM + row] = nn2 - 2.0f * EPSV * sn;
    ws[2 * BDIM + row] = np2 - 2.0f * EPSV * sp;
    ws[3 * BDIM + row] = sqrtf(dap2);
    ws[4 * BDIM + row] = sqrtf(dan2);
    ((unsigned*)ws)[5 * BDIM + row] = 0u;           // hardN = 0.0f (max acc)
    ((unsigned*)ws)[6 * BDIM + row] = 0x7f800000u;  // hardP = +inf (min acc)
  }
}

// ---------------------------------------------------------------------------
// Phase 2: fused f32 GEMM (v_wmma_f32_16x16x4_f32) + masked hard mining.
// Block tile 128(M) x 256(N), K-tile 32.
// 8 waves; each wave owns a 64x64 region = 4x4 tiles of 16x16
// (16 WMMA per K=4 step on 4 A-frags + 4 B-frags).
// Latency hiding via global_prefetch of the next K-tile (zero VGPR cost)
// instead of register staging (which spilled at this acc footprint).
// Grid: (32 row-strips of 128, 8 col-strips of 512, z: 0=neg/max 1=pos/min)
// ---------------------------------------------------------------------------
#define BM 128
#define BN 256
#define BK 32
#define LPAD 4   // stride 36 floats = 144B: 16B-aligned f4 stores, no bank conflicts
#define KTILES (DDIM / BK)

__global__ __launch_bounds__(256) void hardmine_kernel(
    const float* __restrict__ A, const float* __restrict__ Nmat,
    const float* __restrict__ Pmat, float* __restrict__ ws) {
  __shared__ float As[BM][BK + LPAD];   // 18.0 KB
  __shared__ float Bs[BN][BK + LPAD];   // 36.0 KB

  const int which = blockIdx.z;  // 0: negatives (max), 1: positives (min)
  const float* Bmat = which ? Pmat : Nmat;
  const float* rowTerm = ws;
  const float* colTerm = ws + (which ? 2 : 1) * BDIM;
  const float* thrArr  = ws + (which ? 4 : 3) * BDIM;
  unsigned* hard = (unsigned*)ws + (which ? 6 : 5) * BDIM;

  const int i0 = blockIdx.x * BM;
  const int jstrip = blockIdx.y * 512;

  const int tid = threadIdx.x;
  const int wv = tid >> 5;
  const int lane = tid & 31;
  const int wm = wv & 1;   // 0..1 -> 64-row chunk
  const int wn = wv >> 1;  // 0..3 -> 64-col chunk
  const int hi = lane >> 4;
  const int l16 = lane & 15;

  for (int jt = 0; jt < 2; ++jt) {
    const int j0 = jstrip + jt * BN;

    v8f acc[4][4];
#pragma unroll
    for (int mb = 0; mb < 4; ++mb)
#pragma unroll
      for (int nb = 0; nb < 4; ++nb) {
        v8f z = {};
        acc[mb][nb] = z;
      }

    // warm the cache for K-tile 0 (one 128B line per tile row)
    if (tid < BM) __builtin_prefetch(A + (size_t)(i0 + tid) * DDIM, 0, 3);
    __builtin_prefetch(Bmat + (size_t)(j0 + tid) * DDIM, 0, 3);

#pragma unroll 1
    for (int kt = 0; kt < KTILES; ++kt) {
      const int k0 = kt * BK;
      __syncthreads();  // previous compute done -> LDS reusable
      // fill LDS (short-lived temps; loads should hit WGP$/L2 via prefetch)
#pragma unroll
      for (int it = 0; it < 4; ++it) {
        int idx = tid + 256 * it;
        int r = idx >> 3, c = idx & 7;
        float4 v = *(const float4*)(A + (size_t)(i0 + r) * DDIM + k0 + c * 4);
        *(float4*)&As[r][c * 4] = v;
      }
#pragma unroll
      for (int it = 0; it < 8; ++it) {
        int idx = tid + 256 * it;
        int r = idx >> 3, c = idx & 7;
        float4 v = *(const float4*)(Bmat + (size_t)(j0 + r) * DDIM + k0 + c * 4);
        *(float4*)&Bs[r][c * 4] = v;
      }
      __syncthreads();

      // prefetch the next K-tile; flies during the WMMA block below
      if (kt + 1 < KTILES) {
        const int kn = k0 + BK;
        if (tid < BM) __builtin_prefetch(A + (size_t)(i0 + tid) * DDIM + kn, 0, 3);
        __builtin_prefetch(Bmat + (size_t)(j0 + tid) * DDIM + kn, 0, 3);
      }

      // 8 x K=4 steps, 16 WMMA each
#pragma unroll
      for (int kk = 0; kk < BK; kk += 4) {
        const int kb = kk + hi * 2;  // lane<16: K={kk,kk+1}; lane>=16: K={kk+2,kk+3}
        v2f af[4], bf[4];
#pragma unroll
        for (int mb = 0; mb < 4; ++mb)
          af[mb] = *(const v2f*)&As[wm * 64 + mb * 16 + l16][kb];
#pragma unroll
        for (int nb = 0; nb < 4; ++nb)
          bf[nb] = *(const v2f*)&Bs[wn * 64 + nb * 16 + l16][kb];
#pragma unroll
        for (int mb = 0; mb < 4; ++mb)
#pragma unroll
          for (int nb = 0; nb < 4; ++nb)
            acc[mb][nb] = __builtin_amdgcn_wmma_f32_16x16x4_f32(
                false, af[mb], false, bf[nb], (short)0, acc[mb][nb], false, false);
      }
    }

    // epilogue: dot -> distance -> strict mask -> row-wise max/min
    float cT[4];
#pragma unroll
    for (int nb = 0; nb < 4; ++nb)
      cT[nb] = colTerm[j0 + wn * 64 + nb * 16 + l16];

#pragma unroll
    for (int mb = 0; mb < 4; ++mb) {
#pragma unroll
      for (int v = 0; v < 8; ++v) {
        const int m = i0 + wm * 64 + mb * 16 + hi * 8 + v;  // tile M = v + 8*hi
        const float rT = rowTerm[m];
        const float thr = thrArr[m];
        float red = which ? INFINITY : 0.0f;
#pragma unroll
        for (int nb = 0; nb < 4; ++nb) {
          float d2 = rT + cT[nb] - 2.0f * acc[mb][nb][v];
          float dist = sqrtf(fmaxf(d2, 0.0f));
          if (!which) {
            if (dist < thr) red = fmaxf(red, dist);
          } else {
            if (dist > thr) red = fminf(red, dist);
          }
        }
        // reduce across the 16 lanes of this half-wave (same row m)
#pragma unroll
        for (int s = 8; s >= 1; s >>= 1) {
          float o = __shfl_xor(red, s, 32);
          red = which ? fminf(red, o) : fmaxf(red, o);
        }
        if (l16 == 0) {
          // distances >= 0 -> float bits are order-preserving as uint
          if (!which)
            atomicMax(&hard[m], __float_as_uint(red));
          else
            atomicMin(&hard[m], __float_as_uint(red));
        }
      }
    }
  }
}

// ---------------------------------------------------------------------------
// Phase 3: scalar loss.
// ---------------------------------------------------------------------------
__global__ __launch_bounds__(256) void finalize_kernel(
    const float* __restrict__ ws, float* __restrict__ out) {
  __shared__ float s1[256];
  __shared__ float s2[256];
  const float* hardN = ws + 5 * BDIM;
  const float* hardP = ws + 6 * BDIM;
  const float* dap = ws + 3 * BDIM;
  int t = threadIdx.x;

  float accN = 0.f, accP = 0.f;
  for (int i = t; i < BDIM; i += 256) {
    accN += hardN[i];
    float hp = hardP[i];
    accP += (hp < 3.0e38f) ? hp : 0.0f;  // empty mask (+inf) -> 0
  }
  s1[t] = accN;
  s2[t] = accP;
  __syncthreads();
  for (int s = 128; s >= 1; s >>= 1) {
    if (t < s) {
      s1[t] += s1[t + s];
      s2[t] += s2[t + s];
    }
    __syncthreads();
  }
  float d_an = s1[0] / (float)BDIM;
  float d_pp = s2[0] / (float)BDIM;
  __syncthreads();

  float accL = 0.f;
  for (int i = t; i < BDIM; i += 256) {
    float v = dap[i] - 0.5f * d_pp - 0.5f * d_an + 1.0f;
    accL += fmaxf(v, 0.0f);
  }
  s1[t] = accL;
  __syncthreads();
  for (int s = 128; s >= 1; s >>= 1) {
    if (t < s) s1[t] += s1[t + s];
    __syncthreads();
  }
  if (t == 0) out[0] = s1[0] / (float)BDIM;
}

extern "C" void kernel_launch(void* const* d_in, const int* in_sizes, int n_in,
                              void* d_out, int out_size, void* d_ws, size_t ws_size,
                              hipStream_t stream) {
  const float* A = (const float*)d_in[0];  // anchor   [4096,1024]
  const float* P = (const float*)d_in[1];  // positive [4096,1024]
  const float* N = (const float*)d_in[2];  // negative [4096,1024]
  float* ws = (float*)d_ws;
  float* out = (float*)d_out;

  rowstats_kernel<<<BDIM / 8, 256, 0, stream>>>(A, P, N, ws);

  dim3 grid(BDIM / BM, BDIM / 512, 2);  // 32 x 8 x 2
  hardmine_kernel<<<grid, 256, 0, stream>>>(A, N, P, ws);

  finalize_kernel<<<1, 256, 0, stream>>>(ws, out);
}